// criterion_diag_65695819760237
// MI455X (gfx1250) — compile-verified
//
#include <hip/hip_runtime.h>

typedef __attribute__((ext_vector_type(2))) float v2f;
typedef __attribute__((ext_vector_type(8))) float v8f;

#define NCLS 256
#define NDF  512
#define NB   1024
#define KTOT (2 * NDF)       // 1024: concat [X|Y] along K
#define ALP  0.1f
#define HALP 0.05f           // ALP/2

// ---------------------------------------------------------------------------
// Kernel 1: faithful sequential per-class running mean/var.
// One block per class (c = blockIdx.x), one thread per feature d.
// gt[] staged in LDS; (gt[b]==c) is block-uniform -> scalar branch.
// ---------------------------------------------------------------------------
__global__ __launch_bounds__(NDF) void stats_kernel(
    const float* __restrict__ df, const int* __restrict__ gt,
    float* __restrict__ var_out)
{
    __shared__ int gtS[NB];
    const int c = blockIdx.x;
    const int d = threadIdx.x;
    gtS[d]        = gt[d];
    gtS[d + NDF]  = gt[d + NDF];
    __syncthreads();

    float m = 0.0f, v = 0.0f;
    int   n = 0;
    for (int b = 0; b < NB; ++b) {
        if (gtS[b] == c) {                      // block-uniform branch
            const float f  = df[(size_t)b * NDF + d];
            const float nf = (float)n;
            const float c1 = nf / (nf + 1.0f);
            const float c2 = 1.0f / (nf + 1.0f);
            m = (n == 0) ? f : (m * c1 + f * c2);
            const float a = f - m;
            v = v * c1 + a * a * c1 * c2;
            ++n;
        }
    }
    var_out[(size_t)c * NDF + d] = v;
}

// ---------------------------------------------------------------------------
// Kernel 2: per-sample A-matrix rows [X|Y] and per-row constant.
//   X[b,d] = df - ALP*Fc*Var        Y[b,d] = ALP/2 * Var
//   const[b] = -<Fc,df> + ALP/2 * <Fc^2,Var>    (LDS tree reduction)
// ---------------------------------------------------------------------------
__global__ __launch_bounds__(NDF) void prep_a_kernel(
    const float* __restrict__ df, const int* __restrict__ gt,
    const float* __restrict__ fc, const float* __restrict__ var_in,
    float* __restrict__ XY, float* __restrict__ constb)
{
    __shared__ float red[NDF];
    const int b = blockIdx.x;
    const int d = threadIdx.x;
    const int t = gt[b];

    const float fcv = fc[(size_t)t * NDF + d];
    const float va  = var_in[(size_t)t * NDF + d];
    const float dfv = df[(size_t)b * NDF + d];

    XY[(size_t)b * KTOT + d]       = dfv - ALP * fcv * va;   // X
    XY[(size_t)b * KTOT + NDF + d] = HALP * va;              // Y

    red[d] = -fcv * dfv + HALP * fcv * fcv * va;
    __syncthreads();
    for (int s = NDF / 2; s > 0; s >>= 1) {
        if (d < s) red[d] += red[d + s];
        __syncthreads();
    }
    if (d == 0) constb[b] = red[0];
}

// ---------------------------------------------------------------------------
// Kernel 3: B-matrix rows [P|Q] = [fc | fc^2], 256 x 1024, row-major.
// ---------------------------------------------------------------------------
__global__ __launch_bounds__(NDF) void prep_b_kernel(
    const float* __restrict__ fc, float* __restrict__ PQ)
{
    const int c = blockIdx.x;
    const int d = threadIdx.x;
    const float v = fc[(size_t)c * NDF + d];
    PQ[(size_t)c * KTOT + d]       = v;
    PQ[(size_t)c * KTOT + NDF + d] = v * v;
}

// ---------------------------------------------------------------------------
// Kernel 4: f32 WMMA GEMM, 16x64 strip per wave (4 N-tiles).
// A fragment loaded once per K-step and reused across 4 WMMAs:
// 1.25 b64 loads per v_wmma_f32_16x16x4_f32 instead of 2.0.
// B-tile row offsets (16*KTOT floats = 64KB) fold into the 24-bit IOFFSET.
// ---------------------------------------------------------------------------
__global__ __launch_bounds__(32) void wmma_gemm_kernel(
    const float* __restrict__ XY, const float* __restrict__ PQ,
    const float* __restrict__ constb, float* __restrict__ out)
{
    const int tm   = blockIdx.x;          // 0..63  (M tile of 16)
    const int ts   = blockIdx.y;          // 0..3   (N strip of 64)
    const int lane = threadIdx.x;         // wave32
    const int r    = lane & 15;           // M row (A) / N col (B)
    const int kh   = (lane >> 4) << 1;    // K sub-offset: 0 or 2

    const float* Arow = XY + (size_t)(tm * 16 + r) * KTOT + kh;
    const float* Brow = PQ + (size_t)(ts * 64 + r) * KTOT + kh;

    v8f acc0 = {}, acc1 = {}, acc2 = {}, acc3 = {};
#pragma unroll 4
    for (int k = 0; k < KTOT; k += 4) {
        v2f a  = *(const v2f*)(Arow + k);                  // shared A fragment
        v2f b0 = *(const v2f*)(Brow + k);                  // N-tile 0
        v2f b1 = *(const v2f*)(Brow + 16 * KTOT + k);      // N-tile 1
        v2f b2 = *(const v2f*)(Brow + 32 * KTOT + k);      // N-tile 2
        v2f b3 = *(const v2f*)(Brow + 48 * KTOT + k);      // N-tile 3
        acc0 = __builtin_amdgcn_wmma_f32_16x16x4_f32(false, a, false, b0,
                                                     (short)0, acc0, false, false);
        acc1 = __builtin_amdgcn_wmma_f32_16x16x4_f32(false, a, false, b1,
                                                     (short)0, acc1, false, false);
        acc2 = __builtin_amdgcn_wmma_f32_16x16x4_f32(false, a, false, b2,
                                                     (short)0, acc2, false, false);
        acc3 = __builtin_amdgcn_wmma_f32_16x16x4_f32(false, a, false, b3,
                                                     (short)0, acc3, false, false);
    }

    // C/D layout: VGPR j holds M = j (+8 for upper half-wave), N = lane&15.
    const int mbase = tm * 16 + ((lane >> 4) << 3);
    float cst[8];
#pragma unroll
    for (int j = 0; j < 8; ++j) cst[j] = constb[mbase + j];   // -> b128 loads

    const int colbase = ts * 64 + r;
#pragma unroll
    for (int j = 0; j < 8; ++j) {
        const size_t rowoff = (size_t)(mbase + j) * NCLS;
        out[rowoff + colbase +  0] = acc0[j] + cst[j];
        out[rowoff + colbase + 16] = acc1[j] + cst[j];
        out[rowoff + colbase + 32] = acc2[j] + cst[j];
        out[rowoff + colbase + 48] = acc3[j] + cst[j];
    }
}

// ---------------------------------------------------------------------------
extern "C" void kernel_launch(void* const* d_in, const int* in_sizes, int n_in,
                              void* d_out, int out_size, void* d_ws, size_t ws_size,
                              hipStream_t stream) {
    const float* df = (const float*)d_in[0];   // (B, NDF)
    const int*   gt = (const int*)d_in[1];     // (B,)
    const float* fc = (const float*)d_in[2];   // (NCLS, NDF)
    float*       out = (float*)d_out;          // (B, NCLS, 1) flat

    float* var    = (float*)d_ws;                       // NCLS*NDF
    float* XY     = var + (size_t)NCLS * NDF;           // NB*KTOT
    float* PQ     = XY + (size_t)NB * KTOT;             // NCLS*KTOT
    float* constb = PQ + (size_t)NCLS * KTOT;           // NB

    stats_kernel<<<NCLS, NDF, 0, stream>>>(df, gt, var);
    prep_a_kernel<<<NB, NDF, 0, stream>>>(df, gt, fc, var, XY, constb);
    prep_b_kernel<<<NCLS, NDF, 0, stream>>>(fc, PQ);
    wmma_gemm_kernel<<<dim3(NB / 16, NCLS / 16 / 4), 32, 0, stream>>>(XY, PQ, constb, out);
}